// YOLOV5_29317446763195
// MI455X (gfx1250) — compile-verified
//
#include <hip/hip_runtime.h>
#include <math.h>

typedef __attribute__((ext_vector_type(2))) float v2f;
typedef __attribute__((ext_vector_type(8))) float v8f;

#define NC        80
#define NCH       85
#define NT        400
#define NA        3
#define NL        3
#define CAP       8192
#define RECF      12
#define NBLK      256
#define G_OFF     0.5f
#define ANCHOR_Tf 2.91f
#define EPSF      1e-12f

// workspace layout (float indices)
#define TOBJ0     0         // 16*3*80*80 = 307200
#define TOBJ1     307200    // 76800
#define TOBJ2     384000    // 19200
#define CNT_OFF   403200    // 3 ints (match counters)
#define ACC_OFF   403204    // lbox_sum[3], lcls_sum[3]
#define PART_OFF  403216    // 3 * NBLK block partials for lobj
#define REC_OFF   404224    // 3 * CAP * RECF match records
#define ZERO_N    404224    // floats to zero each call

__device__ __forceinline__ float softplus_neg(float x) { // log(1+exp(-x)), stable
    return (x > 0.f) ? log1pf(expf(-x)) : (-x + log1pf(expf(x)));
}
__device__ __forceinline__ float sigmoidf(float x) { return 1.f / (1.f + expf(-x)); }

// ---------------- zero workspace ----------------
__global__ void k_zero(float* __restrict__ ws, int n) {
    int i = blockIdx.x * blockDim.x + threadIdx.x;
    if (i < n) ws[i] = 0.f;
}

// ---------------- build_targets ----------------
__global__ void k_build(const float* __restrict__ targets,
                        const float* __restrict__ anchors,
                        float* __restrict__ ws) {
    int tid = blockIdx.x * blockDim.x + threadIdx.x;
    if (tid >= NL * NA * NT) return;
    int lvl = tid / (NA * NT);
    int a   = (tid / NT) % NA;
    int t   = tid % NT;
    const int Wg[3] = {80, 40, 20};
    float Wf = (float)Wg[lvl], Hf = Wf;

    const float* tr = targets + t * 6;
    float b  = tr[0], c = tr[1];
    float gx = tr[2] * Wf, gy = tr[3] * Hf;
    float gw = tr[4] * Wf, gh = tr[5] * Hf;
    float aw = anchors[(lvl * NA + a) * 2 + 0];
    float ah = anchors[(lvl * NA + a) * 2 + 1];

    float r1 = gw / aw, r2 = gh / ah;
    float m = fmaxf(fmaxf(r1, 1.f / r1), fmaxf(r2, 1.f / r2));
    if (!(m < ANCHOR_Tf)) return;

    float gxi = Wf - gx, gyi = Hf - gy;
    bool pass[5];
    pass[0] = true;
    pass[1] = ((gx  - floorf(gx))  < G_OFF) && (gx  > 1.f);
    pass[2] = ((gy  - floorf(gy))  < G_OFF) && (gy  > 1.f);
    pass[3] = ((gxi - floorf(gxi)) < G_OFF) && (gxi > 1.f);
    pass[4] = ((gyi - floorf(gyi)) < G_OFF) && (gyi > 1.f);
    const float offx[5] = {0.f,  G_OFF, 0.f,  -G_OFF, 0.f};
    const float offy[5] = {0.f,  0.f,  G_OFF, 0.f,  -G_OFF};

    int* cnt = (int*)(ws + CNT_OFF);
    for (int o = 0; o < 5; o++) {
        if (!pass[o]) continue;
        int gi = (int)(gx - offx[o]);   // truncation == floor (positive)
        int gj = (int)(gy - offy[o]);
        int slot = atomicAdd(&cnt[lvl], 1);
        if (slot >= CAP) continue;
        float* r = ws + REC_OFF + (lvl * CAP + slot) * RECF;
        r[0] = b;            r[1] = (float)a;  r[2] = (float)gj; r[3] = (float)gi;
        r[4] = c;            r[5] = gx - (float)gi; r[6] = gy - (float)gj;
        r[7] = gw;           r[8] = gh;        r[9] = aw;        r[10] = ah; r[11] = 0.f;
    }
}

// ---------------- per-match: CIoU + tobj scatter + class BCE ----------------
__global__ void k_match(const float* __restrict__ p0, const float* __restrict__ p1,
                        const float* __restrict__ p2, float* __restrict__ ws) {
    int tid = blockIdx.x * blockDim.x + threadIdx.x;
    int lvl = tid / CAP;
    int m   = tid % CAP;
    if (lvl >= NL) return;
    const int* cnt = (const int*)(ws + CNT_OFF);
    int n = cnt[lvl]; if (n > CAP) n = CAP;
    if (m >= n) return;

    const float* r = ws + REC_OFF + (lvl * CAP + m) * RECF;
    int b = (int)r[0], a = (int)r[1], gj = (int)r[2], gi = (int)r[3], c = (int)r[4];
    float tbx = r[5], tby = r[6], tbw = r[7], tbh = r[8], aw = r[9], ah = r[10];

    const int Wg[3] = {80, 40, 20};
    int Wd = Wg[lvl];
    const float* pi = (lvl == 0) ? p0 : ((lvl == 1) ? p1 : p2);
    long cell = ((long)(b * NA + a) * Wd + gj) * Wd + gi;
    const float* ps = pi + cell * NCH;

    float px = sigmoidf(ps[0]) * 2.f - 0.5f;
    float py = sigmoidf(ps[1]) * 2.f - 0.5f;
    float sw = sigmoidf(ps[2]) * 2.f; float pw = sw * sw * aw;
    float sh = sigmoidf(ps[3]) * 2.f; float ph = sh * sh * ah;

    // CIoU(box1=pred, box2=target)
    float b1x1 = px - pw * 0.5f, b1x2 = px + pw * 0.5f + EPSF;
    float b1y1 = py - ph * 0.5f, b1y2 = py + ph * 0.5f + EPSF;
    float b2x1 = tbx - tbw * 0.5f, b2x2 = tbx + tbw * 0.5f + EPSF;
    float b2y1 = tby - tbh * 0.5f, b2y2 = tby + tbh * 0.5f + EPSF;
    float iw = fmaxf(fminf(b1x2, b2x2) - fmaxf(b1x1, b2x1), 0.f);
    float ih = fmaxf(fminf(b1y2, b2y2) - fmaxf(b1y1, b2y1), 0.f);
    float inter = iw * ih;
    float w1 = b1x2 - b1x1, h1 = b1y2 - b1y1;
    float w2 = b2x2 - b2x1, h2 = b2y2 - b2y1;
    float uni = w1 * h1 + w2 * h2 - inter;
    float iou = inter / uni;
    float cw  = fmaxf(b1x2, b2x2) - fminf(b1x1, b2x1);
    float chh = fmaxf(b1y2, b2y2) - fminf(b1y1, b2y1);
    float c2  = cw * cw + chh * chh;
    float dx  = b2x1 + b2x2 - b1x1 - b1x2;
    float dy  = b2y1 + b2y2 - b1y1 - b1y2;
    float rho2 = dx * dx * 0.25f + dy * dy * 0.25f;
    const float FOUR_OVER_PI2 = 0.405284734569351f; // 4/pi^2
    float dv = atanf(w2 / h2) - atanf(w1 / h1);
    float v  = FOUR_OVER_PI2 * dv * dv;
    float alpha = v / (1.f + EPSF - iou + v);
    float ciou  = iou - (rho2 / c2 + v * alpha);

    float* acc = ws + ACC_OFF;
    atomicAdd(&acc[lvl], 1.f - ciou);

    const int tOff[3] = {TOBJ0, TOBJ1, TOBJ2};
    ws[tOff[lvl] + cell] = fmaxf(ciou, 0.f);   // .set semantics: plain store

    // class BCE (pos_weight = 0.631), one-hot target at c
    const float pwc = 0.631f;
    float cls = 0.f;
    for (int j = 0; j < NC; j++) {
        float x  = ps[5 + j];
        float tt = (j == c) ? 1.f : 0.f;
        cls += (1.f - tt) * x + (1.f + (pwc - 1.f) * tt) * softplus_neg(x);
    }
    atomicAdd(&acc[3 + lvl], cls);
}

// ---------------- objectness BCE streaming reduction ----------------
__global__ void k_obj(const float* __restrict__ pi, const float* __restrict__ tobj,
                      float* __restrict__ part, int ncells) {
    __shared__ float red[8];
    int tid = blockIdx.x * blockDim.x + threadIdx.x;
    int stride = gridDim.x * blockDim.x;
    const float pwv = 0.911f;
    float acc = 0.f;
    for (long cell = tid; cell < ncells; cell += stride) {
        __builtin_prefetch(pi + (cell + stride) * NCH + 4, 0, 1); // global_prefetch_b8
        float x = pi[cell * NCH + 4];
        float t = tobj[cell];
        acc += (1.f - t) * x + (1.f + (pwv - 1.f) * t) * softplus_neg(x);
    }
    for (int off = 16; off >= 1; off >>= 1) acc += __shfl_down(acc, off, 32);
    int lane = threadIdx.x & 31, wid = threadIdx.x >> 5;
    if (lane == 0) red[wid] = acc;
    __syncthreads();
    if (threadIdx.x == 0) {
        float s = 0.f;
        int nw = (int)(blockDim.x >> 5);
        for (int w = 0; w < nw; w++) s += red[w];
        part[blockIdx.x] = s;
    }
}

// ---------------- finalize: WMMA f32 reduction + loss assembly ----------------
__global__ void k_final(const float* __restrict__ ws, float* __restrict__ out) {
    const float* part = ws + PART_OFF;
    const float* acc  = ws + ACC_OFF;
    const int*   cnt  = (const int*)(ws + CNT_OFF);
    int lane = threadIdx.x;

    float S[3];
    for (int l = 0; l < 3; l++) {
        v8f c = {0.f, 0.f, 0.f, 0.f, 0.f, 0.f, 0.f, 0.f};
        const float* p = part + l * NBLK;
        // D = ones(16x4) x B(4x16) + C  ==> every D row = column sums of B.
        // B layout (wave32): lanes 0-15 hold rows K=0 (VGPR0) / K=1 (VGPR1),
        // lanes 16-31 hold rows K=2 / K=3.  64 floats summed per WMMA.
        for (int chunk = 0; chunk < NBLK / 64; chunk++) {
            const float* pb = p + chunk * 64;
            int i0 = (lane < 16) ? lane : lane + 16;
            v2f a = {1.f, 1.f};
            v2f b = {pb[i0], pb[i0 + 16]};
            c = __builtin_amdgcn_wmma_f32_16x16x4_f32(
                    false, a, false, b, (short)0, c, false, false);
        }
        float s = c[0];                              // D[0, lane] for lanes 0-15
        for (int off = 8; off >= 1; off >>= 1) s += __shfl_down(s, off, 16);
        s = __shfl(s, 0, 32);
        S[l] = s;
    }

    if (lane == 0) {
        const float balance[3] = {4.f, 1.f, 0.4f};
        const float ncell[3]   = {307200.f, 76800.f, 19200.f};
        float lobj = 0.f, lbox = 0.f, lcls = 0.f;
        for (int l = 0; l < 3; l++) {
            lobj += (S[l] / ncell[l]) * balance[l];
            int n = cnt[l]; if (n > CAP) n = CAP;
            if (n > 0) {
                lbox += acc[l] / (float)n;
                lcls += acc[3 + l] / ((float)n * (float)NC);
            }
        }
        // s = 3/nl = 1
        lbox *= 0.0296f; lobj *= 0.301f; lcls *= 0.243f;
        float loss = lbox + lobj + lcls;
        out[0] = loss * 16.f;   // loss * bs
        out[1] = lbox; out[2] = lobj; out[3] = lcls; out[4] = loss;
    }
}

extern "C" void kernel_launch(void* const* d_in, const int* in_sizes, int n_in,
                              void* d_out, int out_size, void* d_ws, size_t ws_size,
                              hipStream_t stream) {
    const float* p0 = (const float*)d_in[0];
    const float* p1 = (const float*)d_in[1];
    const float* p2 = (const float*)d_in[2];
    const float* tg = (const float*)d_in[3];
    const float* an = (const float*)d_in[4];
    float* ws  = (float*)d_ws;
    float* out = (float*)d_out;

    k_zero<<<(ZERO_N + 255) / 256, 256, 0, stream>>>(ws, ZERO_N);
    k_build<<<(NL * NA * NT + 255) / 256, 256, 0, stream>>>(tg, an, ws);
    k_match<<<(NL * CAP) / 256, 256, 0, stream>>>(p0, p1, p2, ws);
    k_obj<<<NBLK, 256, 0, stream>>>(p0, ws + TOBJ0, ws + PART_OFF + 0 * NBLK, 307200);
    k_obj<<<NBLK, 256, 0, stream>>>(p1, ws + TOBJ1, ws + PART_OFF + 1 * NBLK, 76800);
    k_obj<<<NBLK, 256, 0, stream>>>(p2, ws + TOBJ2, ws + PART_OFF + 2 * NBLK, 19200);
    k_final<<<1, 32, 0, stream>>>(ws, out);
}